// DynamicsNet_86500641342131
// MI455X (gfx1250) — compile-verified
//
#include <hip/hip_runtime.h>
#include <math.h>

typedef __attribute__((ext_vector_type(2))) float v2f;
typedef __attribute__((ext_vector_type(8))) float v8f;

#define WAVES    4              // waves per block, 16 batches each -> 64 batches/block
#define HSTRIDE  132            // dwords per channel row in h LDS (128 data + 4 pad)
#define HWAVE    (32 * HSTRIDE) // 4224 dwords per wave
#define BFRAGS   (8 * 2 * 8)    // d * ntile * kchunk
#define BSZ      (BFRAGS * 64)  // 8192 dwords of staged Wg fragments
#define SMEM_DW  (WAVES * HWAVE + BSZ)

// ---- compile-time Cayley signs (Cl(3,0), blade index = bitmask) ----
constexpr int popc_c(int x) { int c = 0; while (x) { c += x & 1; x >>= 1; } return c; }
constexpr float csign(int a, int b) {
  int s = 0;
  for (int t = a >> 1; t; t >>= 1) s += popc_c(t & b);
  return (s & 1) ? -1.0f : 1.0f;
}

// rotor q (components at blade indices {0,3,5,6}) times full multivector s
__device__ __forceinline__ void rot_left(const float q[4], const float* s, float* out) {
  constexpr int QI[4] = {0, 3, 5, 6};
#pragma unroll
  for (int k = 0; k < 8; ++k) {
    float a = 0.0f;
#pragma unroll
    for (int i = 0; i < 4; ++i) a += csign(QI[i], QI[i] ^ k) * q[i] * s[QI[i] ^ k];
    out[k] = a;
  }
}

// full multivector t times rotor q
__device__ __forceinline__ void rot_right(const float q[4], const float* t, float* out) {
  constexpr int QI[4] = {0, 3, 5, 6};
#pragma unroll
  for (int m = 0; m < 8; ++m) {
    float a = 0.0f;
#pragma unroll
    for (int i = 0; i < 4; ++i) a += csign(m ^ QI[i], QI[i]) * t[m ^ QI[i]] * q[i];
    out[m] = a;
  }
}

__device__ __forceinline__ void rotor_exp4(float bx, float by, float bz, float q[4]) {
  float th = sqrtf(bx * bx + by * by + bz * bz);
  float sc = (th > 1e-6f) ? (sinf(th) / fmaxf(th, 1e-6f)) : 1.0f;
  q[0] = cosf(th);
  q[1] = sc * bx;
  q[2] = sc * by;
  q[3] = sc * bz;
}

__global__ __launch_bounds__(WAVES * 32) void dynnet_kernel(
    const float* __restrict__ state, const int* __restrict__ action,
    const float* __restrict__ abv, const float* __restrict__ nscale,
    const float* __restrict__ rbv, const float* __restrict__ linW,
    const float* __restrict__ linb, const float* __restrict__ rW,
    const float* __restrict__ rb, float* __restrict__ out_reward,
    float* __restrict__ out_next) {
  extern __shared__ float smem[];
  float* s_h = smem;                 // [WAVES * HWAVE]  per-wave h tiles (A-frag layout)
  float* s_b = smem + WAVES * HWAVE; // [BSZ]            Wg B-fragments

  const int tid  = threadIdx.x;
  const int lane = tid & 31;
  const int wave = tid >> 5;
  const int lo   = lane & 15;
  const int hi   = lane >> 4;
  const int wbase = (blockIdx.x * WAVES + wave) * 16;

  // ---- stage Wg = lin_W[:,:,grade(d)] into LDS, already in B-fragment layout ----
  // fragment f = d*16 + nt*8 + kc holds B[k][n] = Wg[o = nt*16+n][i = kc*4+k][d]
  // element (lane l, vgpr j) at s_b[f*64 + l*2 + j], with n=l%16, k=2*(l/16)+j
  for (int e = tid; e < BSZ; e += WAVES * 32) {
    int f = e >> 6, r = e & 63;
    int l = r >> 1, j = r & 1;
    int d = f >> 4, nt = (f >> 3) & 1, kc = f & 7;
    int o = nt * 16 + (l & 15);
    int i = kc * 4 + 2 * (l >> 4) + j;
    int g = __popc(d); // grade of blade d
    s_b[e] = linW[(o * 32 + i) * 4 + g];
  }

  // ---- per-lane channel rotor (lane == channel) ----
  float qc[4], qcr[4];
  rotor_exp4(-0.5f * rbv[lane * 3 + 0], -0.5f * rbv[lane * 3 + 1],
             -0.5f * rbv[lane * 3 + 2], qc);
  qcr[0] = qc[0]; qcr[1] = -qc[1]; qcr[2] = -qc[2]; qcr[3] = -qc[3];
  const float ns = nscale[lane];

  // ---- phase 1: elementwise pipeline; lane = channel, iteration = batch ----
  for (int it = 0; it < 16; ++it) {
    int b = wbase + it;
    int a = action[b]; // uniform across wave -> scalar load
    float q[4], qr[4];
    rotor_exp4(-0.5f * abv[a * 3 + 0], -0.5f * abv[a * 3 + 1],
               -0.5f * abv[a * 3 + 2], q);
    qr[0] = q[0]; qr[1] = -q[1]; qr[2] = -q[2]; qr[3] = -q[3];

    const float4* sp = reinterpret_cast<const float4*>(state + (size_t)(b * 32 + lane) * 8);
    float4 s0 = sp[0], s1 = sp[1];
    float s[8] = {s0.x, s0.y, s0.z, s0.w, s1.x, s1.y, s1.z, s1.w};

    float t[8], x[8];
    rot_left(q, s, t);
    rot_right(qr, t, x);

    float nn = 1e-6f;
#pragma unroll
    for (int k = 0; k < 8; ++k) nn += x[k] * x[k];
    float inv = ns / sqrtf(nn);
    float h[8];
#pragma unroll
    for (int k = 0; k < 8; ++k) h[k] = x[k] * inv;
    float gate = 0.5f * (1.0f + erff(h[0] * 0.70710678118654752f));
#pragma unroll
    for (int k = 0; k < 8; ++k) h[k] *= gate;

    float t2[8], h2[8];
    rot_left(qc, h, t2);
    rot_right(qcr, t2, h2);

    // h tile in A-fragment-friendly layout: s_h[wave][i=channel][b_off*8 + d]
    float* hp = &s_h[wave * HWAVE + lane * HSTRIDE + it * 8];
    *reinterpret_cast<float4*>(hp + 0) = make_float4(h2[0], h2[1], h2[2], h2[3]);
    *reinterpret_cast<float4*>(hp + 4) = make_float4(h2[4], h2[5], h2[6], h2[7]);

    // stage x into the output buffer (stays L2-resident, re-read in phase 3)
    float4* xp = reinterpret_cast<float4*>(out_next + (size_t)(b * 32 + lane) * 8);
    xp[0] = make_float4(x[0], x[1], x[2], x[3]);
    xp[1] = make_float4(x[4], x[5], x[6], x[7]);
  }

  __syncthreads(); // Wg fragments visible to all waves
  // same-wave global store -> load ordering for the x round-trip
  asm volatile("s_wait_storecnt 0" ::: "memory");

  // ---- phase 3: WMMA channel mix, residual, reward ----
  const float rwb = rb[0];
  float racc[8];
#pragma unroll
  for (int r = 0; r < 8; ++r) racc[r] = 0.0f;

  const float* hb = &s_h[wave * HWAVE];

  for (int nt = 0; nt < 2; ++nt) {
    v8f acc[8];
#pragma unroll
    for (int d = 0; d < 8; ++d) {
      v8f c = {0.f, 0.f, 0.f, 0.f, 0.f, 0.f, 0.f, 0.f};
#pragma unroll
      for (int kc = 0; kc < 8; ++kc) {
        int i0 = kc * 4 + 2 * hi;
        // A frag: lane l -> row m=l%16, K = 2*(l/16)+j
        float a0 = hb[(i0 + 0) * HSTRIDE + lo * 8 + d];
        float a1 = hb[(i0 + 1) * HSTRIDE + lo * 8 + d];
        v2f A = {a0, a1};
        const float* bp = &s_b[(d * 16 + nt * 8 + kc) * 64 + lane * 2];
        v2f B = {bp[0], bp[1]};
        c = __builtin_amdgcn_wmma_f32_16x16x4_f32(
            /*neg_a=*/false, A, /*neg_b=*/false, B,
            /*c_mod=*/(short)0, c, /*reuse_a=*/false, /*reuse_b=*/false);
      }
      acc[d] = c;
    }

    float bias = linb[nt * 16 + lo];
    float rwl  = rW[nt * 16 + lo];
#pragma unroll
    for (int r = 0; r < 8; ++r) {
      int bb = wbase + 8 * hi + r;   // D row M = r + 8*(lane/16)
      int cc = nt * 16 + lo;         // D col N = lane%16
      float4* np = reinterpret_cast<float4*>(out_next + (size_t)(bb * 32 + cc) * 8);
      float4 x0 = np[0], x1 = np[1];
      float n0 = x0.x + acc[0][r] + bias;
      float n1 = x0.y + acc[1][r];
      float n2 = x0.z + acc[2][r];
      float n3 = x0.w + acc[3][r];
      float n4 = x1.x + acc[4][r];
      float n5 = x1.y + acc[5][r];
      float n6 = x1.z + acc[6][r];
      float n7 = x1.w + acc[7][r];
      np[0] = make_float4(n0, n1, n2, n3);
      np[1] = make_float4(n4, n5, n6, n7);
      racc[r] += n0 * rwl;
    }
  }

  // reward: sum over the 16 lanes of each half (wave32 xor-shuffle tree)
#pragma unroll
  for (int r = 0; r < 8; ++r) {
    float v = racc[r];
    v += __shfl_xor(v, 1);
    v += __shfl_xor(v, 2);
    v += __shfl_xor(v, 4);
    v += __shfl_xor(v, 8);
    if (lo == 0) out_reward[wbase + 8 * hi + r] = v + rwb;
  }
}

extern "C" void kernel_launch(void* const* d_in, const int* in_sizes, int n_in,
                              void* d_out, int out_size, void* d_ws, size_t ws_size,
                              hipStream_t stream) {
  const float* state  = (const float*)d_in[0];
  const int*   action = (const int*)d_in[1];
  const float* abv    = (const float*)d_in[2];
  const float* nscale = (const float*)d_in[3];
  const float* rbv    = (const float*)d_in[4];
  const float* linW   = (const float*)d_in[5];
  const float* linb   = (const float*)d_in[6];
  const float* rW     = (const float*)d_in[7];
  const float* rb     = (const float*)d_in[8];

  const int B = in_sizes[1]; // action has one entry per batch
  float* out_reward = (float*)d_out;
  float* out_next   = (float*)d_out + B;

  const int blocks = B / (WAVES * 16);
  const size_t smem_bytes = (size_t)SMEM_DW * sizeof(float);
  dynnet_kernel<<<blocks, WAVES * 32, smem_bytes, stream>>>(
      state, action, abv, nscale, rbv, linW, linb, rW, rb, out_reward, out_next);
}